// LikelihoodMSE_39891656245797
// MI455X (gfx1250) — compile-verified
//
#include <hip/hip_runtime.h>
#include <hip/hip_bf16.h>

// CDNA5 / gfx1250, wave32.
// Batched (B=4096) N=128 SPD: cov += I; left-looking blocked Cholesky in LDS
// with fp32 WMMA (v_wmma_f32_16x16x4_f32) column updates; async global->LDS
// fill; quad = ||L^-1 (u-target)||^2 ; logdet = 2*sum(log diag L).

typedef __attribute__((ext_vector_type(2))) float v2f;
typedef __attribute__((ext_vector_type(8))) float v8f;

#define NDIM 128
#define LDA  129            // padded LDS stride (129 % 64 == 1 -> conflict-free cols)
#define NB   8              // 8 blocks of 16
#define BS   16
#define EPSV 1.0f

__global__ __launch_bounds__(256)
void chol_loss_kernel(const float* __restrict__ u,
                      const float* __restrict__ cov,
                      const float* __restrict__ target,
                      float* __restrict__ ws)
{
    __shared__ float ldsA[NDIM * LDA];
    __shared__ float ldsY[NDIM];
    __shared__ float ldsRed[256];

    const int b    = blockIdx.x;
    const int tid  = threadIdx.x;
    const int wave = tid >> 5;
    const int lane = tid & 31;

    // ---- Async fill: cov[b] (64 KB) -> LDS, 16 B/lane/op, ASYNCcnt-tracked ----
    {
        const float* covb = cov + (size_t)b * NDIM * NDIM;
        #pragma unroll
        for (int k = 0; k < 16; ++k) {
            int i   = tid + k * 256;          // 0..4095 16-byte chunks
            int e   = i << 2;                 // first element of chunk
            int row = e >> 7;
            int col = e & 127;
            unsigned ldsOff = (unsigned)(uintptr_t)(&ldsA[row * LDA + col]);
            unsigned gOff   = (unsigned)(e * 4);   // byte offset from covb
            asm volatile("global_load_async_to_lds_b128 %0, %1, %2"
                         :: "v"(ldsOff), "v"(gOff), "s"(covb)
                         : "memory");
        }
        asm volatile("s_wait_asynccnt 0" ::: "memory");
    }
    __syncthreads();
    if (tid < NDIM) ldsA[tid * LDA + tid] += EPSV;   // cov += eps*I
    __syncthreads();

    const int half = lane >> 4;               // 0|1
    const int l16  = lane & 15;

    // ---- Left-looking blocked Cholesky (block size 16) ----
    for (int kb = 0; kb < NB; ++kb) {
        const int k0 = kb * BS;

        // 1) Update block-column kb with WMMA:
        //    C(ti,kb) -= sum_{p<kb} L(ti,p) * L(kb,p)^T, ti = kb..7.
        //    One tile per wave; WMMA chain accumulates in registers,
        //    single C load/store per tile. Wave-uniform => EXEC all ones.
        if (kb > 0) {
            for (int ti = kb + wave; ti < NB; ti += 8) {
                const int crow0 = ti * BS + half * 8;   // C/D: row = v + 8*half
                const int ccol  = k0 + l16;             //      col = l16
                v8f c;
                #pragma unroll
                for (int v = 0; v < 8; ++v) c[v] = ldsA[(crow0 + v) * LDA + ccol];

                const int am = ti * BS + l16;           // A fragment row (M)
                const int bn = k0 + l16;                // B fragment col (N)
                for (int p = 0; p < kb; ++p) {
                    const int p0 = p * BS;
                    #pragma unroll
                    for (int kk = 0; kk < 4; ++kk) {    // K = 16 as 4 x (K=4)
                        const int kc = p0 + kk * 4 + half * 2;
                        v2f a, bf;
                        a[0]  = -ldsA[am * LDA + kc];   // negate A => C - A*B^T
                        a[1]  = -ldsA[am * LDA + kc + 1];
                        bf[0] =  ldsA[bn * LDA + kc];
                        bf[1] =  ldsA[bn * LDA + kc + 1];
                        c = __builtin_amdgcn_wmma_f32_16x16x4_f32(
                                false, a, false, bf, (short)0, c, false, false);
                    }
                }
                #pragma unroll
                for (int v = 0; v < 8; ++v) ldsA[(crow0 + v) * LDA + ccol] = c[v];
            }
            __syncthreads();
        }

        // 2) Factor diagonal block in place (lower triangle).
        #pragma unroll 1
        for (int j = 0; j < BS; ++j) {
            if (tid == 0) {
                int d = k0 + j;
                ldsA[d * LDA + d] = sqrtf(ldsA[d * LDA + d]);
            }
            __syncthreads();
            if (tid > j && tid < BS) {
                int r = k0 + tid, d = k0 + j;
                ldsA[r * LDA + d] /= ldsA[d * LDA + d];
            }
            __syncthreads();
            if (tid > j && tid < BS) {
                int r = k0 + tid;
                float lrj = ldsA[r * LDA + k0 + j];
                for (int m = j + 1; m <= tid; ++m)
                    ldsA[r * LDA + k0 + m] -= lrj * ldsA[(k0 + m) * LDA + k0 + j];
            }
            __syncthreads();
        }

        // 3) Panel TRSM: rows below solve against L(kb,kb)^T, one row/thread.
        int rowsBelow = NDIM - BS * (kb + 1);
        if (tid < rowsBelow) {
            int r = BS * (kb + 1) + tid;
            float x[BS];
            #pragma unroll
            for (int j = 0; j < BS; ++j) {
                float s = ldsA[r * LDA + k0 + j];
                for (int m = 0; m < j; ++m)
                    s -= x[m] * ldsA[(k0 + j) * LDA + k0 + m];
                x[j] = s / ldsA[(k0 + j) * LDA + k0 + j];
            }
            #pragma unroll
            for (int j = 0; j < BS; ++j) ldsA[r * LDA + k0 + j] = x[j];
        }
        __syncthreads();
    }

    // ---- temp = u - target ----
    if (tid < NDIM)
        ldsY[tid] = u[(size_t)b * NDIM + tid] - target[(size_t)b * NDIM + tid];

    // ---- logdet = 2 * sum(log diag L) ----
    float part = (tid < NDIM) ? logf(ldsA[tid * LDA + tid]) : 0.0f;
    ldsRed[tid] = part;
    __syncthreads();
    for (int s = 128; s > 0; s >>= 1) {
        if (tid < s) ldsRed[tid] += ldsRed[tid + s];
        __syncthreads();
    }
    float logdet2 = 0.0f;
    if (tid == 0) logdet2 = 2.0f * ldsRed[0];
    __syncthreads();

    // ---- Forward solve L y = temp (blocked) ----
    for (int jb = 0; jb < NB; ++jb) {
        if (tid == 0) {
            #pragma unroll
            for (int j = 0; j < BS; ++j) {
                int r = jb * BS + j;
                float s = ldsY[r];
                for (int m = 0; m < j; ++m)
                    s -= ldsA[r * LDA + jb * BS + m] * ldsY[jb * BS + m];
                ldsY[r] = s / ldsA[r * LDA + r];
            }
        }
        __syncthreads();
        int rows = NDIM - BS * (jb + 1);
        if (tid < rows) {
            int r = BS * (jb + 1) + tid;
            float s = ldsY[r];
            #pragma unroll
            for (int m = 0; m < BS; ++m)
                s -= ldsA[r * LDA + jb * BS + m] * ldsY[jb * BS + m];
            ldsY[r] = s;
        }
        __syncthreads();
    }

    // ---- quad = y . y ----
    float q = 0.0f;
    if (tid < NDIM) { float y = ldsY[tid]; q = y * y; }
    ldsRed[tid] = q;
    __syncthreads();
    for (int s = 128; s > 0; s >>= 1) {
        if (tid < s) ldsRed[tid] += ldsRed[tid + s];
        __syncthreads();
    }
    if (tid == 0) ws[b] = ldsRed[0] + logdet2;   // GAMMA = 1
}

// Deterministic fixed-order mean over the 4096 per-batch losses.
__global__ __launch_bounds__(256)
void reduce_mean_kernel(const float* __restrict__ ws, float* __restrict__ out, int n)
{
    __shared__ float red[256];
    float s = 0.0f;
    for (int i = threadIdx.x; i < n; i += 256) s += ws[i];
    red[threadIdx.x] = s;
    __syncthreads();
    for (int k = 128; k > 0; k >>= 1) {
        if (threadIdx.x < k) red[threadIdx.x] += red[threadIdx.x + k];
        __syncthreads();
    }
    if (threadIdx.x == 0) out[0] = red[0] / (float)n;
}

extern "C" void kernel_launch(void* const* d_in, const int* in_sizes, int n_in,
                              void* d_out, int out_size, void* d_ws, size_t ws_size,
                              hipStream_t stream)
{
    const float* u      = (const float*)d_in[0];   // [B, N]
    const float* cov    = (const float*)d_in[1];   // [B, N, N]
    const float* target = (const float*)d_in[2];   // [B, N]
    float* out = (float*)d_out;
    float* ws  = (float*)d_ws;                     // B floats of scratch

    const int B = in_sizes[0] / NDIM;              // 4096

    chol_loss_kernel<<<B, 256, 0, stream>>>(u, cov, target, ws);
    reduce_mean_kernel<<<1, 256, 0, stream>>>(ws, out, B);
}